// MyLayer_31817117729481
// MI455X (gfx1250) — compile-verified
//
#include <hip/hip_runtime.h>

// CDNA5 / gfx1250. fp32 WMMA path: V_WMMA_F32_16X16X4_F32.
typedef float v2f __attribute__((ext_vector_type(2)));
typedef float v8f __attribute__((ext_vector_type(8)));

#define NN   256   // matrix dim
#define NC   64    // channels
#define NH   128   // N/2 (output tile dims)

// ---------------------------------------------------------------------------
// Kernel 1: H2 = haar @ haar, keep only columns 0..127, and store pre-packed
// for WMMA-B fragments:  Bp[(k>>1)*256 + j*2 + (k&1)] = H2[k][j]
// so a lane's B fragment {K=k, K=k+1} at column j is one contiguous float2.
// 256*128 = 32768 outputs, K=256 -> 16.8 MFLOP, negligible.
// ---------------------------------------------------------------------------
__global__ void h2_pack_kernel(const float* __restrict__ haar,
                               float* __restrict__ Bp) {
    int t = blockIdx.x * blockDim.x + threadIdx.x;   // 0..32767
    int k = t >> 7;          // 0..255  (reduction index of the main GEMM)
    int j = t & 127;         // 0..127  (output column)
    const float* hrow = haar + k * NN;
    float s = 0.0f;
    #pragma unroll 4
    for (int m = 0; m < NN; ++m)
        s = fmaf(hrow[m], haar[m * NN + j], s);
    Bp[(k >> 1) * 256 + j * 2 + (k & 1)] = s;
}

// ---------------------------------------------------------------------------
// Kernel 2: one workgroup per (b, c'); 256 threads = 8 wave32.
// Wave w computes output rows i' = 16w..16w+15 (a 16-row M strip) against all
// 8 N-tiles (columns 0..127) with K=256, using v_wmma_f32_16x16x4_f32.
//
// Row gather (from the reference's transpose/reshape/slice):
//   out[b, c', i', j'] = (x[b, c] @ H2)[i, j']   with c = i'&63, i = 4c'+(i'>>6)
//
// H2 (pre-packed) is staged into LDS in two 64-kpair chunks. LDS rows are
// padded to 288 floats so lane halves 0-15 / 16-31 (adjacent k-pair rows)
// hit disjoint bank sets (row stride % 64 dwords == 32).
// ---------------------------------------------------------------------------
#define KPAD 288                  // 256 data floats + 32 pad floats per k-pair row
__global__ __launch_bounds__(256)
void wmma_xh2_kernel(const float* __restrict__ x,
                     const float* __restrict__ Bp,
                     float* __restrict__ out) {
    __shared__ float Bs[64 * KPAD];                 // 73,728 B LDS

    const int wg  = blockIdx.x;                     // b*64 + c'
    const int b   = wg >> 6;
    const int cp  = wg & 63;                        // c'
    const int tid = threadIdx.x;
    const int w   = tid >> 5;                       // wave id 0..7 (M strip)
    const int l   = tid & 31;                       // lane
    const int ml  = l & 15;                         // M (A) / N (B,C) within tile
    const int h   = l >> 4;                         // lane half: K subgroup

    // This lane's A-matrix row (output row i' = 16w + ml)
    const int ip = 16 * w + ml;                     // i' in [0,128)
    const int ch = ip & 63;                         // source channel
    const int ib = 4 * cp + (ip >> 6);              // source row in x[b,ch]
    const float* xrow = x + (((size_t)b * NC + ch) * NN + (size_t)ib) * NN;

    v8f acc[8] = {};                                // 8 N-tiles of 16x16 f32

    for (int ck = 0; ck < 2; ++ck) {
        // ---- stage 64 k-pair rows (k = ck*128 .. ck*128+127) into LDS ----
        const float4* src = (const float4*)(Bp + ck * 16384);
        for (int i4 = tid; i4 < 4096; i4 += 256) {
            int row = i4 >> 6;                      // k-pair row 0..63
            int co4 = i4 & 63;                      // float4 within row
            ((float4*)(Bs + row * KPAD))[co4] = src[i4];
        }
        __syncthreads();

        const int kbase = ck * 128;
        // prefetch this lane's next x chunk while computing this one
        if (ck == 0) __builtin_prefetch(xrow + 128, 0, 0);

        #pragma unroll 2
        for (int ks = 0; ks < 128; ks += 4) {
            // A fragment: lanes 0-15 carry K = ks..ks+1, lanes 16-31 K = ks+2..ks+3
            v2f a = *(const v2f*)(xrow + kbase + ks + 2 * h);
            const float* brow = Bs + ((ks >> 1) + h) * KPAD + ml * 2;
            #pragma unroll
            for (int n = 0; n < 8; ++n) {
                v2f bf = *(const v2f*)(brow + n * 32);   // col = 16n + ml
                acc[n] = __builtin_amdgcn_wmma_f32_16x16x4_f32(
                    /*neg_a=*/false, a, /*neg_b=*/false, bf,
                    /*c_mod=*/(short)0, acc[n],
                    /*reuse_a=*/false, /*reuse_b=*/false);
            }
        }
        __syncthreads();
    }

    // ---- store: C/D layout: VGPR r, lanes 0-15 -> M=r, lanes 16-31 -> M=r+8 ----
    float* obase = out + ((size_t)wg * NH + (size_t)(16 * w)) * NH;
    #pragma unroll
    for (int n = 0; n < 8; ++n) {
        #pragma unroll
        for (int r = 0; r < 8; ++r) {
            obase[(size_t)(r + 8 * h) * NH + 16 * n + ml] = acc[n][r];
        }
    }
}

// ---------------------------------------------------------------------------
// Launch: inputs per reference order: d_in[0] = x (fp32, 16*64*256*256),
// d_in[1] = haar (fp32, 256*256). Output: fp32, 16*64*128*128.
// Workspace: 128 KB for packed H2.
// ---------------------------------------------------------------------------
extern "C" void kernel_launch(void* const* d_in, const int* in_sizes, int n_in,
                              void* d_out, int out_size, void* d_ws, size_t ws_size,
                              hipStream_t stream) {
    const float* x    = (const float*)d_in[0];
    const float* haar = (const float*)d_in[1];
    float* out = (float*)d_out;
    float* Bp  = (float*)d_ws;                      // 256*128 floats = 128 KB

    // Kernel 1: pack H2 (256x128) into WMMA-B friendly layout.
    h2_pack_kernel<<<dim3(128), dim3(256), 0, stream>>>(haar, Bp);

    // Kernel 2: 16*64 = 1024 workgroups, one per (b, c').
    wmma_xh2_kernel<<<dim3(16 * NC), dim3(256), 0, stream>>>(x, Bp, out);
}